// CPMLoss_5746666242356
// MI455X (gfx1250) — compile-verified
//
#include <hip/hip_runtime.h>
#include <math.h>

// ---------------------------------------------------------------------------
// CPM loss forward for MI455X (gfx1250, wave32).
//   inputs: [4*128*16, 4096] f32, targets (layout known -> unused), P=128
//   output: single f32 scalar
// Gram matrices c_m @ c_m^T computed with V_WMMA_F32_16X16X4_F32 (f32 WMMA,
// keeps full precision for the sqrt/min chain). Everything else is trivially
// memory-bound (128 MiB read ~= 5.5us at 23.3 TB/s).
// ---------------------------------------------------------------------------

#define D_DIM   4096
#define P_IDS   128
#define K_SAMP  16
#define NMOD    4
#define MARGIN_ 0.2f

typedef __attribute__((ext_vector_type(2))) float v2f;
typedef __attribute__((ext_vector_type(8))) float v8f;

// ---- 1) centers: mean over K=16 samples. 2M output floats, float4 per thread.
__global__ void cpm_centers_kernel(const float* __restrict__ in,
                                   float* __restrict__ centers) {
    int t  = blockIdx.x * blockDim.x + threadIdx.x;   // 524288 threads
    int c  = t >> 10;                                  // center index 0..511
    int d4 = (t & 1023) << 2;                          // dim offset (x4)
    const float4* base = (const float4*)(in + (size_t)c * K_SAMP * D_DIM + d4);
    float4 s = make_float4(0.f, 0.f, 0.f, 0.f);
#pragma unroll
    for (int k = 0; k < K_SAMP; ++k) {
        float4 v = base[k * (D_DIM / 4)];
        s.x += v.x; s.y += v.y; s.z += v.z; s.w += v.w;
    }
    const float inv = 1.0f / (float)K_SAMP;
    float4 o = make_float4(s.x * inv, s.y * inv, s.z * inv, s.w * inv);
    *(float4*)(centers + (size_t)c * D_DIM + d4) = o;
}

// ---- 2) Gram per modality via f32 WMMA. One wave (32 threads) per 16x16 tile.
// grid = 256 blocks: [m(2b) | tile_i(3b) | tile_j(3b)], block = 32 threads.
__global__ void cpm_gram_wmma_kernel(const float* __restrict__ centers,
                                     float* __restrict__ gram) {
    int b    = blockIdx.x;
    int m    = b >> 6;
    int tile = b & 63;
    int ti   = tile >> 3;
    int tj   = tile & 7;
    int lane = threadIdx.x;            // 0..31, wave32
    int r    = lane & 15;
    int koff = (lane >> 4) << 1;       // 0 for lanes 0-15, 2 for lanes 16-31

    // A-tile rows: C_m[ti*16 + r][*] ; B-tile rows: C_m[tj*16 + r][*]
    const float* Arow = centers + (size_t)(m * P_IDS + ti * 16 + r) * D_DIM + koff;
    const float* Brow = centers + (size_t)(m * P_IDS + tj * 16 + r) * D_DIM + koff;

    v8f acc = {0.f, 0.f, 0.f, 0.f, 0.f, 0.f, 0.f, 0.f};
#pragma unroll 8
    for (int kk = 0; kk < D_DIM; kk += 4) {
        v2f a  = *(const v2f*)(Arow + kk);   // 8B-aligned global_load_b64
        v2f bb = *(const v2f*)(Brow + kk);
        // D = A(16x4,f32) x B(4x16,f32) + C ; 8 args (neg_a,A,neg_b,B,c_mod,C,reuseA,reuseB)
        acc = __builtin_amdgcn_wmma_f32_16x16x4_f32(
            false, a, false, bb, (short)0, acc, false, false);
    }

    // C/D layout: VGPR v -> row M = v + (lane>=16 ? 8 : 0), col N = lane%16
    int rowAdd = (lane >> 4) << 3;
    float* gm = gram + (size_t)m * P_IDS * P_IDS;
#pragma unroll
    for (int v = 0; v < 8; ++v) {
        int gi = ti * 16 + v + rowAdd;
        int gj = tj * 16 + (lane & 15);
        gm[gi * P_IDS + gj] = acc[v];
    }
}

// ---- 3) off-diagonal row min of within-modality distances.
// d2[i][j] = G[i][i] + G[j][j] - 2 G[i][j]; one thread per (m,i).
__global__ void cpm_selfmin_kernel(const float* __restrict__ gram,
                                   float* __restrict__ ansel) {
    int t = blockIdx.x * blockDim.x + threadIdx.x;   // 512 threads
    int m = t >> 7;
    int i = t & 127;
    const float* gm = gram + (size_t)m * P_IDS * P_IDS;
    float gii = gm[i * P_IDS + i];
    float mn = INFINITY;
    for (int j = 0; j < P_IDS; ++j) {
        if (j == i) continue;
        float d2 = gii + gm[j * P_IDS + j] - 2.0f * gm[i * P_IDS + j];
        mn = fminf(mn, d2);
    }
    ansel[t] = sqrtf(fmaxf(mn, 1e-12f));
}

// ---- 4) cross-modality diagonal distances; one wave per (pair, identity).
// pairs: 0:(c2,c3)=ap123  1:(c1,c3)=an123  2:(c1,c4)=ap124  3:(c2,c4)=an124
__global__ void cpm_crossdiag_kernel(const float* __restrict__ centers,
                                     float* __restrict__ crossd) {
    int wave = (blockIdx.x * blockDim.x + threadIdx.x) >> 5;  // 0..511
    int lane = threadIdx.x & 31;
    int pair = wave >> 7;
    int i    = wave & 127;
    int pa   = (((pair + 1) & 2) == 0) ? 1 : 0;   // 1,0,0,1
    int pb   = 2 + (pair >> 1);                   // 2,2,3,3
    const float* A = centers + (size_t)(pa * P_IDS + i) * D_DIM;
    const float* B = centers + (size_t)(pb * P_IDS + i) * D_DIM;
    float s = 0.f;
#pragma unroll 4
    for (int it = 0; it < D_DIM / 4 / 32; ++it) {
        int d = (lane + it * 32) * 4;
        float4 a = *(const float4*)(A + d);
        float4 b = *(const float4*)(B + d);
        float dx = a.x - b.x, dy = a.y - b.y, dz = a.z - b.z, dw = a.w - b.w;
        s += dx * dx + dy * dy + dz * dz + dw * dw;
    }
    for (int off = 16; off > 0; off >>= 1)
        s += __shfl_down(s, off, 32);
    if (lane == 0)
        crossd[wave] = sqrtf(fmaxf(s, 1e-12f));
}

// ---- 5) six margin-ranking means -> scalar. One block of 128 threads.
__global__ void cpm_final_kernel(const float* __restrict__ crossd,
                                 const float* __restrict__ ansel,
                                 float* __restrict__ out) {
    __shared__ float sh[128];
    int i = threadIdx.x;
    float ap123 = crossd[0 * 128 + i];
    float an123 = crossd[1 * 128 + i];
    float ap124 = crossd[2 * 128 + i];
    float an124 = crossd[3 * 128 + i];
    float an11  = ansel[0 * 128 + i];
    float an22  = ansel[1 * 128 + i];
    float an33  = ansel[2 * 128 + i];
    float an44  = ansel[3 * 128 + i];
    float s = fmaxf(ap123 - an123 + MARGIN_, 0.f)
            + 0.5f * (fmaxf(ap123 - an33 + MARGIN_, 0.f) +
                      fmaxf(ap123 - an11 + MARGIN_, 0.f))
            + fmaxf(ap124 - an124 + MARGIN_, 0.f)
            + 0.5f * (fmaxf(ap124 - an44 + MARGIN_, 0.f) +
                      fmaxf(ap124 - an22 + MARGIN_, 0.f));
    sh[i] = s;
    __syncthreads();
    for (int off = 64; off > 0; off >>= 1) {
        if (i < off) sh[i] += sh[i + off];
        __syncthreads();
    }
    if (i == 0) out[0] = sh[0] * (1.0f / (float)P_IDS) * 0.5f;
}

extern "C" void kernel_launch(void* const* d_in, const int* in_sizes, int n_in,
                              void* d_out, int out_size, void* d_ws, size_t ws_size,
                              hipStream_t stream) {
    (void)in_sizes; (void)n_in; (void)out_size; (void)ws_size;
    const float* in = (const float*)d_in[0];   // [8192, 4096] f32
    float* ws      = (float*)d_ws;
    float* centers = ws;                                   // 4*128*4096 = 2M f32
    float* gram    = centers + NMOD * P_IDS * D_DIM;       // 4*128*128  = 64K f32
    float* crossd  = gram + NMOD * P_IDS * P_IDS;          // 512 f32
    float* ansel   = crossd + NMOD * P_IDS;                // 512 f32
    float* out     = (float*)d_out;

    // 524288 threads for 2M center floats (float4 each)
    cpm_centers_kernel<<<2048, 256, 0, stream>>>(in, centers);
    // 256 tiles, one wave32 per 16x16 tile (f32 WMMA)
    cpm_gram_wmma_kernel<<<256, 32, 0, stream>>>(centers, gram);
    // 512 waves for the 4x128 cross-modal diagonal distances
    cpm_crossdiag_kernel<<<128, 128, 0, stream>>>(centers, crossd);
    // 512 threads for the 4x128 off-diagonal mins
    cpm_selfmin_kernel<<<2, 256, 0, stream>>>(gram, ansel);
    // final scalar
    cpm_final_kernel<<<1, 128, 0, stream>>>(crossd, ansel, out);
}